// E2M1WeightQuantizer_635655160010
// MI455X (gfx1250) — compile-verified
//
#include <hip/hip_runtime.h>
#include <hip/hip_bf16.h>
#include <stdint.h>

// MX-FP4 (E2M1) block quantizer for MI455X (gfx1250).
// Memory-bound streaming kernel (~264 MB read + ~528 MB written -> ~34us at
// 23.3 TB/s). Uses the CDNA5 async-tensor path: global_load_async_to_lds_b128
// (ASYNCcnt) in a 4-deep software pipeline with s_wait_asynccnt/s_wait_dscnt,
// wave32 8-lane shuffle reductions for the per-block amax, and non-temporal
// cache hints on all bulk traffic (every byte is single-use; keeps the 192 MB
// L2 from write-allocating dead output lines against the inbound stream).

typedef __attribute__((ext_vector_type(4))) float v4f;

#define TPB    256
#define DEPTH  4                 // async pipeline depth (stages in flight)
#define WAIT_ASYNC_3 "s_wait_asynccnt 3"   // DEPTH-1

static_assert(DEPTH == 4, "wait-string hardcodes DEPTH-1 == 3");

__device__ __forceinline__ float groupmax8(float x) {
    // max over groups of 8 lanes (xor 1,2,4 stays inside an 8-lane group)
    x = fmaxf(x, __shfl_xor(x, 1, 32));
    x = fmaxf(x, __shfl_xor(x, 2, 32));
    x = fmaxf(x, __shfl_xor(x, 4, 32));
    return x;
}

__device__ __forceinline__ void encode1(float v, float inv_scale, float scale,
                                        float eps, float& enc_out, float& deq_out) {
    float wn = v * inv_scale;                       // w_norm
    float a  = fmaxf(fabsf(wn) + eps, 0.0f);        // abs_shifted
    // sign_bit = floor((2-sign)/2): 0 for v>0, 1 for v<=0
    float sbit = (v > 0.0f) ? 0.0f : 1.0f;
    float sgn  = (v > 0.0f) ? 1.0f : -1.0f;
    // threshold-count encode against E2M1 bounds
    float c0 = (a > 0.25f) ? 1.0f : 0.0f;
    float c1 = (a > 0.75f) ? 1.0f : 0.0f;
    float c2 = (a > 1.25f) ? 1.0f : 0.0f;
    float c3 = (a > 1.75f) ? 1.0f : 0.0f;
    float c4 = (a > 2.50f) ? 1.0f : 0.0f;
    float c5 = (a > 3.50f) ? 1.0f : 0.0f;
    float c6 = (a > 5.00f) ? 1.0f : 0.0f;
    float ord = ((c0 + c1) + (c2 + c3)) + ((c4 + c5) + c6);
    enc_out = sbit * 8.0f + ord;
    // mantissa = E2M1_VALUES[ord] as telescoping sum of value deltas
    float m = 0.5f * ((c0 + c1) + (c2 + c3)) + (c4 + c5) + 2.0f * c6;
    deq_out = m * scale * sgn;
}

__global__ __launch_bounds__(TPB) void e2m1_quant_kernel(
    const float* __restrict__ w, const float* __restrict__ eps_p,
    float* __restrict__ out, long long N, long long NB,
    long long nchunks, int iters)
{
    __shared__ v4f stage[DEPTH][TPB];
    const int tid = threadIdx.x;
    const long long lastc = nchunks - 1;

    auto chunk_of = [&](long long it) -> long long {
        return (it * (long long)gridDim.x + (long long)blockIdx.x) * TPB + tid;
    };

    auto issue = [&](int slot, long long c) {
        if (c > lastc) c = lastc;   // clamp tail -> harmless dummy load
        const float* src = w + (c << 2);
        unsigned long long ga = (unsigned long long)(uintptr_t)src;
        unsigned lds = (unsigned)(uintptr_t)(&stage[slot][tid]); // LDS byte offset
        // NT: weights are read exactly once -> don't retain in L2
        asm volatile("global_load_async_to_lds_b128 %0, %1, off th:TH_LOAD_NT"
                     :: "v"(lds), "v"(ga) : "memory");
    };

    // ---- prologue: put DEPTH-1 stages in flight (always issue; dummies are
    //      clamped and land in slots that are never read) ----
    #pragma unroll
    for (int s = 0; s < DEPTH - 1; ++s) issue(s, chunk_of(s));

    for (int it = 0; it < iters; ++it) {
        const int slot = it & (DEPTH - 1);

        // Re-fill the pipeline: stage it+DEPTH-1 goes into the slot consumed
        // last iteration. Close the LDS WAR hazard vs. that ds_load first.
        asm volatile("s_wait_dscnt 0" ::: "memory");
        issue((it + DEPTH - 1) & (DEPTH - 1), chunk_of(it + DEPTH - 1));

        // DEPTH async loads outstanding cover stages [it .. it+DEPTH-1].
        // They complete in order, so <= DEPTH-1 outstanding => stage 'it' landed.
        asm volatile(WAIT_ASYNC_3 ::: "memory");

        const long long c = chunk_of(it);
        if (c < nchunks) {
            v4f v = *(volatile v4f*)&stage[slot][tid];

            // per-block (32 elems = 8 lanes x 4) amax via wave shuffles
            float local = fmaxf(fmaxf(fabsf(v.x), fabsf(v.y)),
                                fmaxf(fabsf(v.z), fabsf(v.w)));
            float amax = groupmax8(local);

            // e8m0 scale: exp2(ceil(max(log2(amax/6), -127)))
            float descale = amax * (1.0f / 6.0f);
            float e_f = ceilf(fmaxf(__log2f(descale), -127.0f));
            int e = (int)e_f;
            float scale     = (e >= -126) ? __int_as_float((e + 127) << 23) : 0x1.0p-127f;
            float inv_scale = (e <=  126) ? __int_as_float((127 - e) << 23) : 0x1.0p-127f;

            // learned block offset eps = 0.5*tanh(p)
            const long long blk = c >> 3;
            float eps = 0.5f * tanhf(eps_p[blk]);

            float e0, e1, e2, e3, d0, d1, d2, d3;
            encode1(v.x, inv_scale, scale, eps, e0, d0);
            encode1(v.y, inv_scale, scale, eps, e1, d1);
            encode1(v.z, inv_scale, scale, eps, e2, d2);
            encode1(v.w, inv_scale, scale, eps, e3, d3);
            v4f dq = {d0, d1, d2, d3};
            v4f en = {e0, e1, e2, e3};

            // NT streaming stores: outputs are never re-read by this kernel
            __builtin_nontemporal_store(dq, (v4f*)(out + (c << 2)));      // deq
            __builtin_nontemporal_store(en, (v4f*)(out + N + (c << 2)));  // encoded
            if ((tid & 7) == 0) {
                out[2 * N + blk]      = e_f + 127.0f;  // e8m0_scale_uint8 (as float)
                out[2 * N + NB + blk] = eps;           // eps_base
            }
        }
    }
}

extern "C" void kernel_launch(void* const* d_in, const int* in_sizes, int n_in,
                              void* d_out, int out_size, void* d_ws, size_t ws_size,
                              hipStream_t stream) {
    (void)n_in; (void)out_size; (void)d_ws; (void)ws_size;
    const float* w     = (const float*)d_in[0];   // weight_fp, flat [out_f*in_f]
    const float* eps_p = (const float*)d_in[1];   // eps_param,  flat [out_f*in_f/32]
    float* out = (float*)d_out;

    const long long N  = (long long)in_sizes[0];  // 8192*8192
    const long long NB = (long long)in_sizes[1];  // N/32
    const long long nchunks = N >> 2;             // one float4 chunk per lane-stage

    int grid = 4096;
    long long per_iter = (long long)grid * TPB;
    if (nchunks < per_iter) {
        grid = (int)((nchunks + TPB - 1) / TPB);
        if (grid < 1) grid = 1;
        per_iter = (long long)grid * TPB;
    }
    const int iters = (int)((nchunks + per_iter - 1) / per_iter);

    e2m1_quant_kernel<<<grid, TPB, 0, stream>>>(w, eps_p, out, N, NB, nchunks, iters);
}